// AttnDecoderRNN_54949811585256
// MI455X (gfx1250) — compile-verified
//
#include <hip/hip_runtime.h>
#include <hip/hip_bf16.h>
#include <math.h>

// Sizes from the reference
#define HH 1024
#define EE 1024
#define VV 50257
#define LL 64
#define HE 2048   // H + E
#define G3 3072   // 3*H

typedef __attribute__((ext_vector_type(2))) float v2f;
typedef __attribute__((ext_vector_type(8))) float v8f;
typedef unsigned int u32x4 __attribute__((ext_vector_type(4)));
typedef int i32x8 __attribute__((ext_vector_type(8)));
typedef int i32x4 __attribute__((ext_vector_type(4)));

// ---------------- workspace layout (floats) ----------------
#define WS_XA   0        // xa   [2048] concat(embedded, h0)
#define WS_ALOG 2048     // alog [64]   attention logits
#define WS_XC   2112     // xc   [2048] concat(embedded, attn_applied)
#define WS_XV   4160     // xv   [1024] relu(comb)
#define WS_GI   5184     // gi   [3072]
#define WS_GH   8256     // gh   [3072]
#define WS_HN   11328    // hn   [1024] h_new
#define WS_LG   12352    // lg   [50257] vocab logits
#define WS_RED  62609    // red  [1]    max + log(sum exp)

// ---- TDM-staged WMMA GEMV tiling ----
#define WPB        6                 // waves per block (192 threads)
#define KCHUNK     64                // k-elements staged per TDM transfer
#define ROWSTRIDE  65                // LDS row stride in floats (64 + 1 pad DWORD)
#define TILE_F     (16 * ROWSTRIDE)  // floats per LDS tile buffer
#define TILE_B     (TILE_F * 4)      // bytes per LDS tile buffer (4160)
#define GEMV_SMEM  (WPB * 2 * TILE_B)   // 49,920 B dynamic LDS per block

// Build D# group0 (128b): count=1 | lds_addr@[63:32] | global_addr@[120:64] | type=2@[127:126]
// Build D# group1 (256b): flags | tensor_dim0@[79:48] | tensor_dim1@[111:80] |
//                         tile_dim0@[127:112] | tile_dim1@[143:128] | stride@[207:160]
__device__ __forceinline__ void tdm_load_tile(const float* gsrc, unsigned lds_off,
                                              int rows_valid, int k_stride) {
  union { unsigned long long q[2]; u32x4 v; } g0;
  union { unsigned long long q[4]; i32x8 v; } g1;
  const unsigned long long gaddr = (unsigned long long)(const void*)gsrc;
  g0.q[0] = 1ull | ((unsigned long long)lds_off << 32);
  g0.q[1] = (gaddr & ((1ull << 57) - 1)) | (2ull << 62);
  // flags: data_size=2 (4B) @[17:16], pad_enable @[20], pad_interval=5 (64 DW) @[24:22],
  //        pad_amount=0 (1 DW) @[31:25], workgroup_mask=0
  const unsigned long long flags = (2ull << 16) | (1ull << 20) | (5ull << 22);
  g1.q[0] = flags | ((unsigned long long)(KCHUNK & 0xFFFF) << 48);  // tensor_dim0[15:0]
  g1.q[1] = ((unsigned long long)KCHUNK >> 16)                      // tensor_dim0[31:16]
          | ((unsigned long long)(unsigned)rows_valid << 16)        // tensor_dim1
          | ((unsigned long long)KCHUNK << 48);                     // tile_dim0
  g1.q[2] = 16ull                                                   // tile_dim1 = 16 rows
          | ((unsigned long long)(unsigned)k_stride << 32);         // stride[31:0]
  g1.q[3] = 0;                                                      // stride[47:32] = 0
  const i32x4 z4 = {0, 0, 0, 0};
  const i32x8 z8 = {0, 0, 0, 0, 0, 0, 0, 0};
  __builtin_amdgcn_tensor_load_to_lds(g0.v, g1.v, z4, z4, z8, 0);
}

// ---------------------------------------------------------------------------
// f32 WMMA GEMV:  y[n] = act( sum_k W[n][k] * x[k] + b[n] ),  W: N x K row-major.
// One wave per 16-row tile. TDM DMAs 16 x KCHUNK weight tiles into LDS (double
// buffered, padded rows for bank-conflict-free reads), V_WMMA_F32_16X16X4_F32
// accumulates:  A = weight tile (lane&15 = row, half-wave picks K pair),
// B = x[k] broadcast across columns, D column 0 = the 16 dot products
// (lane 0 -> rows 0..7 in acc[0..7], lane 16 -> rows 8..15).
// Requires K % KCHUNK == 0 (true for all call sites: 1024 / 2048).
// ---------------------------------------------------------------------------
__global__ void gemv_wmma_f32(const float* __restrict__ W,
                              const float* __restrict__ x,
                              const float* __restrict__ b,
                              float* __restrict__ y,
                              int N, int K, int act) {
  extern __shared__ float lds_f[];
  const int lane  = threadIdx.x & 31;
  const int wib   = (int)(threadIdx.x >> 5);
  const int gwave = (int)blockIdx.x * WPB + wib;
  const int numTiles = (N + 15) >> 4;
  if (gwave >= numTiles) return;            // wave-uniform; EXEC stays full

  const int row0  = gwave << 4;
  const int m     = lane & 15;              // A-matrix row owned by this lane
  const int kHalf = (lane >> 4) << 1;       // 0 for lanes 0-15, 2 for 16-31
  int rows_valid = N - row0; if (rows_valid > 16) rows_valid = 16;

  const unsigned lds_base = __builtin_amdgcn_groupstaticsize() + (unsigned)(wib * 2) * TILE_B;
  const float* Wt = W + (size_t)row0 * (size_t)K;
  const int numChunks = K / KCHUNK;

  int buf = 0;
  tdm_load_tile(Wt, lds_base, rows_valid, K);           // prime buffer 0
  v8f acc = {};
  for (int c = 0; c < numChunks; ++c) {
    const int kc = c * KCHUNK;
    if (c + 1 < numChunks) {
      tdm_load_tile(Wt + kc + KCHUNK, lds_base + (unsigned)(buf ^ 1) * TILE_B,
                    rows_valid, K);
      __builtin_amdgcn_s_wait_tensorcnt(1);             // chunk c landed
    } else {
      __builtin_amdgcn_s_wait_tensorcnt(0);
    }
    const float* __restrict__ T =
        lds_f + (size_t)(wib * 2 + buf) * TILE_F + (size_t)m * ROWSTRIDE + kHalf;
    const float* __restrict__ xk = x + kc + kHalf;
#pragma unroll
    for (int k = 0; k < KCHUNK; k += 4) {
      v2f a;  a[0]  = T[k];  a[1]  = T[k + 1];          // A[m][kHalf..+1]
      v2f bx; bx[0] = xk[k]; bx[1] = xk[k + 1];         // B rows bcast over N
      acc = __builtin_amdgcn_wmma_f32_16x16x4_f32(
          false, a, false, bx, (short)0, acc, false, false);
    }
    buf ^= 1;
  }

  // Column N=0 of D: lane 0 holds M=0..7, lane 16 holds M=8..15.
  if ((lane & 15) == 0) {
    const int base = row0 + ((lane >> 4) << 3);
#pragma unroll
    for (int i = 0; i < 8; ++i) {
      const int rr = base + i;
      if (rr < N) {
        float v = acc[i] + (b ? b[rr] : 0.0f);
        if (act == 1) v = fmaxf(v, 0.0f);
        y[rr] = v;
      }
    }
  }
}

// Build xa = concat(emb[input], h0)
__global__ void k_prep_concat0(const int* __restrict__ tok,
                               const float* __restrict__ emb,
                               const float* __restrict__ hidden,
                               float* __restrict__ xa) {
  const int i = blockIdx.x * blockDim.x + threadIdx.x;
  if (i >= HE) return;
  const int t = tok[0];
  xa[i] = (i < EE) ? emb[(size_t)t * EE + i] : hidden[i - EE];
}

// softmax over 64 logits, attn_applied = attn_w @ enc, build xc, emit weights
__global__ void k_attn_apply(const float* __restrict__ alog,
                             const float* __restrict__ enc,   // [L][H]
                             const float* __restrict__ xa,    // embedded in [0,E)
                             float* __restrict__ xc,          // [2048]
                             float* __restrict__ attnw_out) { // d_out + V + H
  __shared__ float w[LL];
  __shared__ float ms[2];
  const int tid = threadIdx.x;
  if (tid == 0) {
    float mx = alog[0];
    for (int l = 1; l < LL; ++l) mx = fmaxf(mx, alog[l]);
    float s = 0.0f;
    for (int l = 0; l < LL; ++l) s += __expf(alog[l] - mx);
    ms[0] = mx; ms[1] = s;
  }
  __syncthreads();
  if (tid < LL) {
    const float wv = __expf(alog[tid] - ms[0]) / ms[1];
    w[tid] = wv;
    attnw_out[tid] = wv;
  }
  __syncthreads();
  for (int j = tid; j < HH; j += blockDim.x) {
    float a = 0.0f;
#pragma unroll 8
    for (int l = 0; l < LL; ++l) a = fmaf(w[l], enc[l * HH + j], a);
    xc[EE + j] = a;
    xc[j] = xa[j];           // embedded half of the concat
  }
}

// GRU gate combine (torch order r,z,n): h_new = (1-z)*n + z*h0
__global__ void k_gru_combine(const float* __restrict__ gi,
                              const float* __restrict__ gh,
                              const float* __restrict__ hidden,
                              float* __restrict__ hn,
                              float* __restrict__ hnew_out) { // d_out + V
  const int i = blockIdx.x * blockDim.x + threadIdx.x;
  if (i >= HH) return;
  const float r = 1.0f / (1.0f + __expf(-(gi[i]          + gh[i])));
  const float z = 1.0f / (1.0f + __expf(-(gi[i + HH]     + gh[i + HH])));
  const float n = tanhf(gi[i + 2 * HH] + r * gh[i + 2 * HH]);
  const float h = (1.0f - z) * n + z * hidden[i];
  hn[i] = h;
  hnew_out[i] = h;
}

// Single-block reduction: red = max(lg) + log(sum exp(lg - max))
__global__ void k_lse_reduce(const float* __restrict__ lg, float* __restrict__ red) {
  __shared__ float s[1024];
  const int tid = threadIdx.x;
  float mx = -INFINITY;
  for (int i = tid; i < VV; i += 1024) mx = fmaxf(mx, lg[i]);
  s[tid] = mx;
  __syncthreads();
  for (int off = 512; off > 0; off >>= 1) {
    if (tid < off) s[tid] = fmaxf(s[tid], s[tid + off]);
    __syncthreads();
  }
  const float M = s[0];
  __syncthreads();
  float sm = 0.0f;
  for (int i = tid; i < VV; i += 1024) sm += __expf(lg[i] - M);
  s[tid] = sm;
  __syncthreads();
  for (int off = 512; off > 0; off >>= 1) {
    if (tid < off) s[tid] += s[tid + off];
    __syncthreads();
  }
  if (tid == 0) red[0] = M + __logf(s[0]);
}

__global__ void k_logsoftmax_write(const float* __restrict__ lg,
                                   const float* __restrict__ red,
                                   float* __restrict__ logp) {
  const int i = blockIdx.x * blockDim.x + threadIdx.x;
  if (i < VV) logp[i] = lg[i] - red[0];
}

static inline int gemv_grid(int N) {
  const int tiles = (N + 15) / 16;          // one wave per 16-row tile
  return (tiles + WPB - 1) / WPB;
}

extern "C" void kernel_launch(void* const* d_in, const int* in_sizes, int n_in,
                              void* d_out, int out_size, void* d_ws, size_t ws_size,
                              hipStream_t stream) {
  (void)in_sizes; (void)n_in; (void)out_size; (void)ws_size;
  const int*   tok    = (const int*)  d_in[0];
  const float* hidden = (const float*)d_in[1];
  const float* enc    = (const float*)d_in[2];
  const float* emb    = (const float*)d_in[3];
  const float* attn_w = (const float*)d_in[4];
  const float* attn_b = (const float*)d_in[5];
  const float* comb_w = (const float*)d_in[6];
  const float* comb_b = (const float*)d_in[7];
  const float* w_ih   = (const float*)d_in[8];
  const float* b_ih   = (const float*)d_in[9];
  const float* w_hh   = (const float*)d_in[10];
  const float* b_hh   = (const float*)d_in[11];
  const float* out_w  = (const float*)d_in[12];
  const float* out_b  = (const float*)d_in[13];

  float* out  = (float*)d_out;            // [logp(V) | h_new(H) | attn_w(L)]
  float* ws   = (float*)d_ws;
  float* xa   = ws + WS_XA;
  float* alog = ws + WS_ALOG;
  float* xc   = ws + WS_XC;
  float* xv   = ws + WS_XV;
  float* gi   = ws + WS_GI;
  float* gh   = ws + WS_GH;
  float* hn   = ws + WS_HN;
  float* lg   = ws + WS_LG;
  float* red  = ws + WS_RED;

  // 1) xa = concat(embedded, h0)
  k_prep_concat0<<<HE / 256, 256, 0, stream>>>(tok, emb, hidden, xa);
  // 2) attn logits = attn_w @ xa + attn_b          (64 x 2048)
  gemv_wmma_f32<<<gemv_grid(LL), WPB * 32, GEMV_SMEM, stream>>>(attn_w, xa, attn_b, alog, LL, HE, 0);
  // 3) softmax + attn_applied + xc; emit attn weights
  k_attn_apply<<<1, 256, 0, stream>>>(alog, enc, xa, xc, out + VV + HH);
  // 4) xv = relu(comb_w @ xc + comb_b)             (1024 x 2048)
  gemv_wmma_f32<<<gemv_grid(EE), WPB * 32, GEMV_SMEM, stream>>>(comb_w, xc, comb_b, xv, EE, HE, 1);
  // 5) gi = w_ih @ xv + b_ih                        (3072 x 1024)
  gemv_wmma_f32<<<gemv_grid(G3), WPB * 32, GEMV_SMEM, stream>>>(w_ih, xv, b_ih, gi, G3, EE, 0);
  // 6) gh = w_hh @ h0 + b_hh                        (3072 x 1024)
  gemv_wmma_f32<<<gemv_grid(G3), WPB * 32, GEMV_SMEM, stream>>>(w_hh, hidden, b_hh, gh, G3, HH, 0);
  // 7) GRU combine -> h_new (ws + d_out)
  k_gru_combine<<<HH / 256, 256, 0, stream>>>(gi, gh, hidden, hn, out + VV);
  // 8) vocab logits = out_w @ h_new + out_b         (50257 x 1024) -- dominant
  gemv_wmma_f32<<<gemv_grid(VV), WPB * 32, GEMV_SMEM, stream>>>(out_w, hn, out_b, lg, VV, HH, 0);
  // 9) log-sum-exp reduction
  k_lse_reduce<<<1, 1024, 0, stream>>>(lg, red);
  // 10) logp = lg - red
  k_logsoftmax_write<<<(VV + 255) / 256, 256, 0, stream>>>(lg, red, out);
}